// FDDiscriminator_46213848105111
// MI455X (gfx1250) — compile-verified
//
#include <hip/hip_runtime.h>
#include <hip/hip_bf16.h>

typedef __attribute__((ext_vector_type(2))) float v2f;
typedef __attribute__((ext_vector_type(8))) float v8f;

#define T_SLICES 128
#define NPTS     512
#define CCH      16
// exp(-d/2) = exp2(-d * log2(e)/2)
#define NEG_HALF_LOG2E (-0.7213475204444817f)

__global__ __launch_bounds__(256) void mmd_slice_kernel(
    const float* __restrict__ x, const float* __restrict__ y,
    float* __restrict__ partial)
{
    __shared__ float sX[NPTS * CCH];   // 32 KB: X slice, point-major (16 contiguous ch)
    __shared__ float sY[NPTS * CCH];   // 32 KB
    __shared__ float snx[NPTS];        // |x_i|^2
    __shared__ float sny[NPTS];        // |y_j|^2
    __shared__ float sred[8][2];

    const int s   = blockIdx.x;        // slice 0..127
    const int tid = threadIdx.x;       // 0..255

    // ---- Stage slice s of x and y into LDS (float4 chunks, 2048 per array) ----
    for (int it = 0; it < 8; ++it) {
        int idx = it * 256 + tid;      // chunk id
        int p = idx >> 2;              // point
        int q = idx & 3;               // which float4 of the 16 channels
        const float4* gx = (const float4*)(x + (size_t)p * (T_SLICES * CCH) + s * CCH + q * 4);
        const float4* gy = (const float4*)(y + (size_t)p * (T_SLICES * CCH) + s * CCH + q * 4);
        float4 vx = *gx;
        float4 vy = *gy;
        *(float4*)(&sX[p * CCH + q * 4]) = vx;
        *(float4*)(&sY[p * CCH + q * 4]) = vy;
    }
    __syncthreads();

    // ---- Squared norms: 512 points / 256 threads ----
    for (int r = 0; r < 2; ++r) {
        int p = r * 256 + tid;
        float ax = 0.f, ay = 0.f;
#pragma unroll
        for (int c = 0; c < CCH; ++c) {
            float vx = sX[p * CCH + c]; ax += vx * vx;
            float vy = sY[p * CCH + c]; ay += vy * vy;
        }
        snx[p] = ax;
        sny[p] = ay;
    }
    __syncthreads();

    const int wave = tid >> 5;
    const int lane = tid & 31;
    const int half = lane >> 4;        // K/M half select
    const int l    = lane & 15;

    float accXX = 0.f, accXY = 0.f;

    for (int strip = 0; strip < 4; ++strip) {
        const int ti   = wave + strip * 8;   // row tile 0..31
        const int irow = ti * 16 + l;

        // A operand (reused across inner loops): a[kk] = X[irow][4*kk + 2*half + {0,1}]
        v2f a[4];
#pragma unroll
        for (int kk = 0; kk < 4; ++kk) {
            const float* pa = &sX[irow * CCH + kk * 4 + 2 * half];
            a[kk] = v2f{pa[0], pa[1]};
        }
        // Row norms for the 8 C/D rows this lane owns: M = v + 8*half
        float nx[8];
#pragma unroll
        for (int v = 0; v < 8; ++v)
            nx[v] = snx[ti * 16 + v + 8 * half];

        // ---- XX tiles ----
        for (int tj = 0; tj < 32; ++tj) {
            const int jcol = tj * 16 + l;
            v8f c = {};
#pragma unroll
            for (int kk = 0; kk < 4; ++kk) {
                const float* pb = &sX[jcol * CCH + kk * 4 + 2 * half];
                v2f b = v2f{pb[0], pb[1]};
                c = __builtin_amdgcn_wmma_f32_16x16x4_f32(
                        false, a[kk], false, b, (short)0, c, false, false);
            }
            const float ny = snx[jcol];
#pragma unroll
            for (int v = 0; v < 8; ++v) {
                float d = nx[v] + ny - 2.0f * c[v];
                accXX += __builtin_amdgcn_exp2f(d * NEG_HALF_LOG2E);
            }
        }

        // ---- XY tiles ----
        for (int tj = 0; tj < 32; ++tj) {
            const int jcol = tj * 16 + l;
            v8f c = {};
#pragma unroll
            for (int kk = 0; kk < 4; ++kk) {
                const float* pb = &sY[jcol * CCH + kk * 4 + 2 * half];
                v2f b = v2f{pb[0], pb[1]};
                c = __builtin_amdgcn_wmma_f32_16x16x4_f32(
                        false, a[kk], false, b, (short)0, c, false, false);
            }
            const float ny = sny[jcol];
#pragma unroll
            for (int v = 0; v < 8; ++v) {
                float d = nx[v] + ny - 2.0f * c[v];
                accXY += __builtin_amdgcn_exp2f(d * NEG_HALF_LOG2E);
            }
        }
    }

    // ---- wave32 tree reduce (deterministic) ----
#pragma unroll
    for (int off = 16; off > 0; off >>= 1) {
        accXX += __shfl_xor(accXX, off, 32);
        accXY += __shfl_xor(accXY, off, 32);
    }
    if (lane == 0) { sred[wave][0] = accXX; sred[wave][1] = accXY; }
    __syncthreads();

    if (tid == 0) {
        float Sxx = 0.f, Sxy = 0.f;
        for (int w = 0; w < 8; ++w) { Sxx += sred[w][0]; Sxy += sred[w][1]; }
        const float wgt = (s == 0 || s == T_SLICES - 1) ? 1.0f : 2.0f;
        const float c1  = 1.0f / (254.0f * 512.0f * 511.0f);
        const float c2  = 1.0f / (254.0f * 512.0f * 512.0f);
        partial[s] = wgt * ((Sxx - 512.0f) * c1 - 2.0f * Sxy * c2);
    }
}

__global__ void mmd_final_kernel(const float* __restrict__ partial,
                                 float* __restrict__ out)
{
    if (threadIdx.x == 0) {
        float r = 0.f;
        for (int s2 = 0; s2 < T_SLICES; ++s2) r += partial[s2];
        out[0] = r;
    }
}

extern "C" void kernel_launch(void* const* d_in, const int* in_sizes, int n_in,
                              void* d_out, int out_size, void* d_ws, size_t ws_size,
                              hipStream_t stream)
{
    (void)in_sizes; (void)n_in; (void)out_size; (void)ws_size;
    const float* x = (const float*)d_in[0];
    const float* y = (const float*)d_in[1];
    float* partial = (float*)d_ws;      // 128 floats
    float* out     = (float*)d_out;     // 1 float

    mmd_slice_kernel<<<T_SLICES, 256, 0, stream>>>(x, y, partial);
    mmd_final_kernel<<<1, 32, 0, stream>>>(partial, out);
}